// PopulationSlow_25168508355298
// MI455X (gfx1250) — compile-verified
//
#include <hip/hip_runtime.h>

typedef __attribute__((ext_vector_type(2))) float v2f;
typedef __attribute__((ext_vector_type(8))) float v8f;

#define NN 8192         // N (matrix dim)
#define KK 16           // rank
#define MT 2            // m-tiles (16 cols each) per wave
#define NT_CHUNK 16     // n-tiles (16 rows each) per wave

// W[n,m] = sum_k gamma_k * v[k,n] * u[k,m]
// gamma_k = 0.9 + 0.0995 * sigmoid(eta_k)
// Implemented as chained V_WMMA_F32_16X16X4_F32 (full fp32, K=16 = 4 steps of K=4).
__global__ __launch_bounds__(32) void wslow_wmma(
    const float* __restrict__ eta,
    const float* __restrict__ v,   // [K][N]
    const float* __restrict__ u,   // [K][N]
    float* __restrict__ W)         // [N][N]
{
    const int lane = threadIdx.x & 31;
    const int half = lane >> 4;     // 0: lanes 0-15, 1: lanes 16-31
    const int l    = lane & 15;

    // Per-lane gammas for this lane's K slots in each of the 4 WMMA steps.
    // Step s, VGPR0 holds K = 4s + 2*half, VGPR1 holds K = 4s + 2*half + 1.
    float gam[8];
#pragma unroll
    for (int s = 0; s < 4; ++s) {
        const int k0 = 4 * s + 2 * half;
        const float e0 = eta[k0];
        const float e1 = eta[k0 + 1];
        gam[2 * s]     = 0.9f + 0.0995f / (1.0f + __expf(-e0));
        gam[2 * s + 1] = 0.9f + 0.0995f / (1.0f + __expf(-e1));
    }

    const int m0 = blockIdx.x * (MT * 16);

    // B fragments (gamma folded in), loaded ONCE per wave.
    // B layout (4x16 fp32, 2 VGPRs): VGPR0 = K {0 | 2}, VGPR1 = K {1 | 3}, col = l.
    v2f b[MT][4];
#pragma unroll
    for (int t = 0; t < MT; ++t) {
#pragma unroll
        for (int s = 0; s < 4; ++s) {
            const int k0 = 4 * s + 2 * half;
            const int m  = m0 + t * 16 + l;
            b[t][s].x = gam[2 * s]     * u[(size_t)k0 * NN + m];
            b[t][s].y = gam[2 * s + 1] * u[(size_t)(k0 + 1) * NN + m];
        }
    }

    int n0 = blockIdx.y * (NT_CHUNK * 16);
    for (int it = 0; it < NT_CHUNK; ++it, n0 += 16) {
        // A fragment (16x4 fp32, 2 VGPRs): same per-lane K mapping as B, row = l.
        v2f a[4];
#pragma unroll
        for (int s = 0; s < 4; ++s) {
            const int k0 = 4 * s + 2 * half;
            a[s].x = v[(size_t)k0 * NN + n0 + l];
            a[s].y = v[(size_t)(k0 + 1) * NN + n0 + l];
        }

#pragma unroll
        for (int t = 0; t < MT; ++t) {
            v8f c = {};
#pragma unroll
            for (int s = 0; s < 4; ++s) {
                // 8 args: (neg_a, A, neg_b, B, c_mod, C, reuse_a, reuse_b)
                c = __builtin_amdgcn_wmma_f32_16x16x4_f32(
                        false, a[s], false, b[t][s], (short)0, c, false, false);
            }
            // D layout (16x16 fp32, 8 VGPRs): VGPR r -> row (r + 8*half), col l.
            const int col = m0 + t * 16 + l;
#pragma unroll
            for (int r = 0; r < 8; ++r) {
                const int row = n0 + r + 8 * half;
                __builtin_nontemporal_store(c[r], &W[(size_t)row * NN + col]);
            }
        }
    }
}

extern "C" void kernel_launch(void* const* d_in, const int* in_sizes, int n_in,
                              void* d_out, int out_size, void* d_ws, size_t ws_size,
                              hipStream_t stream) {
    const float* eta = (const float*)d_in[0];  // [16]
    const float* v   = (const float*)d_in[1];  // [16][8192]
    const float* u   = (const float*)d_in[2];  // [16][8192]
    float* W = (float*)d_out;                  // [8192][8192]

    dim3 block(32, 1, 1);                                  // 1 wave32 per block
    dim3 grid(NN / (MT * 16), NN / (NT_CHUNK * 16), 1);    // 256 x 32 waves
    wslow_wmma<<<grid, block, 0, stream>>>(eta, v, u, W);
}